// ContinuousFlowLoss_20761871909425
// MI455X (gfx1250) — compile-verified
//
#include <hip/hip_runtime.h>

typedef __attribute__((ext_vector_type(2))) float v2f;
typedef __attribute__((ext_vector_type(8))) float v8f;

namespace {
constexpr int kH = 384, kW = 384;
constexpr int kTile = 16, kHalo = 3;
constexpr int kTDim = kTile + 2 * kHalo;   // 22
constexpr int kTSz  = kTDim * kTDim;       // 484
constexpr int kBx = kW / kTile;            // 24
constexpr int kBy = kH / kTile;            // 24
}

__device__ __forceinline__ float tap_weight(float d2) {
  // w = dist/(dist+1), dist = exp(-(sqrt(clip(d2,1e-8)) * 2 - 10))
  float d = sqrtf(fmaxf(d2, 1e-8f));
  float dist = __expf(10.0f - 2.0f * d);
  return dist / (dist + 1.0f);
}

// Sum 256 floats in LDS with V_WMMA_F32_16X16X4_F32 (exact fp32 FMAs,
// fixed hardware order -> deterministic). Must be executed by all 32 lanes
// of a full wave (EXEC all ones). Returns the total in every lane.
// A (16x4 fp32, 2 VGPRs) x B(all-ones) + C  gives row-sums replicated over
// columns; summing the 8 C/D VGPRs per lane then xor-16 combine yields the
// grand total of the 64 A elements; chain 4 accumulating WMMAs for 256.
__device__ __forceinline__ float wmma_sum256(const float* v, int lane) {
  v2f ones; ones[0] = 1.0f; ones[1] = 1.0f;
  v8f c = {0.0f, 0.0f, 0.0f, 0.0f, 0.0f, 0.0f, 0.0f, 0.0f};
#pragma unroll
  for (int t = 0; t < 4; ++t) {
    v2f a;
    a[0] = v[lane + 64 * t];
    a[1] = v[lane + 32 + 64 * t];
    c = __builtin_amdgcn_wmma_f32_16x16x4_f32(false, a, false, ones,
                                              (short)0, c, false, false);
  }
  float s = c[0] + c[1] + c[2] + c[3] + c[4] + c[5] + c[6] + c[7];
  s += __shfl_xor(s, 16, 32);
  return s;
}

__global__ void cfl_init_ws(float* ws) {
  for (int i = 0; i < 16; ++i) ws[i] = 0.0f;
}

// Pass 1: hm0r = fixed-kernel 7x7 conv of hm_0 (zero padded); global max -> norm bits.
__global__ __launch_bounds__(256) void cfl_norm_kernel(const float* __restrict__ hm0,
                                                       unsigned* __restrict__ norm_bits) {
  __shared__ float s_t[kTSz];
  __shared__ float s_kw[49];
  __shared__ float s_red[256];
  const int tx = threadIdx.x, ty = threadIdx.y;
  const int tid = ty * kTile + tx;
  const int b = blockIdx.z;
  const int bx0 = blockIdx.x * kTile, by0 = blockIdx.y * kTile;
  const float* hb = hm0 + (size_t)b * kH * kW;

  for (int t = tid; t < kTSz; t += 256) {
    int r = t / kTDim, c = t % kTDim;
    int gy = by0 - kHalo + r, gx = bx0 - kHalo + c;
    bool in = (gy >= 0) && (gy < kH) && (gx >= 0) && (gx < kW);
    s_t[t] = in ? hb[gy * kW + gx] : 0.0f;
  }
  if (tid < 49) {
    int i = tid / 7, j = tid % 7;
    float di = (float)(i - kHalo), dj = (float)(j - kHalo);
    s_kw[tid] = tap_weight(di * di + dj * dj);
  }
  __syncthreads();

  float acc = 0.0f;
#pragma unroll
  for (int i = 0; i < 7; ++i)
#pragma unroll
    for (int j = 0; j < 7; ++j)
      acc += s_t[(ty + i) * kTDim + (tx + j)] * s_kw[i * 7 + j];

  s_red[tid] = acc;
  __syncthreads();
  for (int st = 128; st >= 1; st >>= 1) {
    if (tid < st) s_red[tid] = fmaxf(s_red[tid], s_red[tid + st]);
    __syncthreads();
  }
  if (tid == 0) atomicMax(norm_bits, __float_as_uint(s_red[0]));
}

// Pass 2: rec losses + consistency + reg per pixel; WMMA block-reduce -> partial.
__global__ __launch_bounds__(256) void cfl_main_kernel(
    const float* __restrict__ hm0, const float* __restrict__ hm1,
    const float* __restrict__ off_f, const float* __restrict__ off_b,
    const unsigned* __restrict__ norm_bits, float* __restrict__ partials) {
  __shared__ float s_hm0[kTSz], s_hm1[kTSz];
  __shared__ float s_fx[kTSz], s_fy[kTSz], s_bx[kTSz], s_by[kTSz];
  __shared__ float s_kw[49];
  __shared__ float s_red[256];

  const int tx = threadIdx.x, ty = threadIdx.y;
  const int tid = ty * kTile + tx;
  const int b = blockIdx.z;
  const int bx0 = blockIdx.x * kTile, by0 = blockIdx.y * kTile;
  const size_t HW = (size_t)kH * kW;

  const float* h0b = hm0 + b * HW;
  const float* h1b = hm1 + b * HW;
  const float* fxp = off_f + b * 2 * HW;
  const float* fyp = fxp + HW;
  const float* bxp = off_b + b * 2 * HW;
  const float* byp = bxp + HW;

  for (int t = tid; t < kTSz; t += 256) {
    int r = t / kTDim, c = t % kTDim;
    int gy = by0 - kHalo + r, gx = bx0 - kHalo + c;
    bool in = (gy >= 0) && (gy < kH) && (gx >= 0) && (gx < kW);
    int gi = gy * kW + gx;
    s_hm0[t] = in ? h0b[gi] : 0.0f;
    s_hm1[t] = in ? h1b[gi] : 0.0f;
    s_fx[t] = in ? fxp[gi] : 0.0f;
    s_fy[t] = in ? fyp[gi] : 0.0f;
    s_bx[t] = in ? bxp[gi] : 0.0f;
    s_by[t] = in ? byp[gi] : 0.0f;
  }
  if (tid < 49) {
    int i = tid / 7, j = tid % 7;
    float di = (float)(i - kHalo), dj = (float)(j - kHalo);
    s_kw[tid] = tap_weight(di * di + dj * dj);
  }
  __syncthreads();

  const int h = by0 + ty, w = bx0 + tx;
  const float fh = (float)h, fw = (float)w;
  float rec1f = 0.0f, rec0b = 0.0f, hm0r = 0.0f, hm1r = 0.0f;
#pragma unroll
  for (int i = 0; i < 7; ++i) {
#pragma unroll
    for (int j = 0; j < 7; ++j) {
      int li = (ty + i) * kTDim + (tx + j);
      float s0 = s_hm0[li], s1 = s_hm1[li];
      float kwv = s_kw[i * 7 + j];
      hm0r += s0 * kwv;
      hm1r += s1 * kwv;
      float sx = fw + (float)(j - kHalo);
      float sy = fh + (float)(i - kHalo);
      float dx = fw - (sx + s_fx[li]);
      float dy = fh - (sy + s_fy[li]);
      rec1f += s0 * tap_weight(dx * dx + dy * dy);
      dx = fw - (sx + s_bx[li]);
      dy = fh - (sy + s_by[li]);
      rec0b += s1 * tap_weight(dx * dx + dy * dy);
    }
  }

  const float normv = fmaxf(__uint_as_float(*norm_bits), 1.0f);
  const float inv = 1.0f / normv;
  float rw = fabsf(hm0r - hm1r) * inv;
  float e1 = (rec1f - hm1r) * inv;
  float e2 = (rec0b - hm0r) * inv;
  float loss = 0.5f * (e1 * e1 + e2 * e2) * (1.0f + rw);

  // consistency (LAM_CONS * 0.5)
  const int lc = (ty + kHalo) * kTDim + (tx + kHalo);
  float fx = s_fx[lc], fy = s_fy[lc], bx = s_bx[lc], by = s_by[lc];
  int gx1 = (int)fminf(fmaxf(rintf(fw + fx), 0.0f), (float)(kW - 1));
  int gy1 = (int)fminf(fmaxf(rintf(fh + fy), 0.0f), (float)(kH - 1));
  float brx = bxp[gy1 * kW + gx1], bry = byp[gy1 * kW + gx1];
  float t1 = fx + brx * ((fabsf(fx) > 0.5f) ? 1.0f : 0.0f);
  float t2 = fy + bry * ((fabsf(fy) > 0.5f) ? 1.0f : 0.0f);
  int gx2 = (int)fminf(fmaxf(rintf(fw + bx), 0.0f), (float)(kW - 1));
  int gy2 = (int)fminf(fmaxf(rintf(fh + by), 0.0f), (float)(kH - 1));
  float frx = fxp[gy2 * kW + gx2], fry = fyp[gy2 * kW + gx2];
  float t3 = bx + frx * ((fabsf(bx) > 0.5f) ? 1.0f : 0.0f);
  float t4 = by + fry * ((fabsf(by) > 0.5f) ? 1.0f : 0.0f);
  loss += 0.5f * (t1 * t1 + t2 * t2 + t3 * t3 + t4 * t4);

  // reg (LAM_REG = 0.01)
  float h0c = s_hm0[lc], h1c = s_hm1[lc];
  loss += 0.01f * (((h0c == 0.0f) ? (fabsf(fx) + fabsf(fy)) : 0.0f) +
                   ((h1c == 0.0f) ? (fabsf(bx) + fabsf(by)) : 0.0f));

  s_red[tid] = loss;
  __syncthreads();
  if (tid < 32) {
    float tot = wmma_sum256(s_red, tid);
    if (tid == 0) {
      int blin = (blockIdx.z * kBy + blockIdx.y) * kBx + blockIdx.x;
      partials[blin] = tot;
    }
  }
}

// Smoothing: one block per (call, b, p); 9x9 masked mean/var of offset.
__global__ __launch_bounds__(128) void cfl_smooth_kernel(
    const float* __restrict__ off_f, const float* __restrict__ off_b,
    const int* __restrict__ pts0, const int* __restrict__ pts1,
    const int* __restrict__ len0, const int* __restrict__ len1,
    int B, int P, float* __restrict__ outp) {
  __shared__ float sA[128], sB[128], sC[128];
  __shared__ float s_mx, s_my, s_n;
  const int idx = blockIdx.x;
  const int per_call = B * P;
  const int call = idx / per_call;
  const int rem = idx % per_call;
  const int b = rem / P, p = rem % P;
  const float* off = (call == 0) ? off_f : off_b;
  const int* pts = (call == 0) ? pts0 : pts1;
  const int* lens = (call == 0) ? len0 : len1;
  const size_t HW = (size_t)kH * kW;

  const int x = pts[(b * P + p) * 2 + 0];
  const int y = pts[(b * P + p) * 2 + 1];
  const int tid = threadIdx.x;

  float vx = 0.0f, vy = 0.0f, m = 0.0f;
  if (tid < 81) {
    int dy = tid / 9 - 4, dx = tid % 9 - 4;
    int iy = y + dy, ix = x + dx;
    m = ((iy >= 0) && (iy < kH) && (ix >= 0) && (ix < kW)) ? 1.0f : 0.0f;
    int IY = min(max(iy, 0), kH - 1);
    int IX = min(max(ix, 0), kW - 1);
    const float* ob = off + b * 2 * HW;
    vx = ob[IY * kW + IX];
    vy = ob[HW + IY * kW + IX];
  }
  sA[tid] = vx * m; sB[tid] = vy * m; sC[tid] = m;
  __syncthreads();
  for (int st = 64; st >= 1; st >>= 1) {
    if (tid < st) {
      sA[tid] += sA[tid + st];
      sB[tid] += sB[tid + st];
      sC[tid] += sC[tid + st];
    }
    __syncthreads();
  }
  if (tid == 0) { s_n = sC[0]; s_mx = sA[0] / sC[0]; s_my = sB[0] / sC[0]; }
  __syncthreads();
  float dxv = vx - s_mx, dyv = vy - s_my;
  sA[tid] = dxv * dxv * m;
  sB[tid] = dyv * dyv * m;
  __syncthreads();
  for (int st = 64; st >= 1; st >>= 1) {
    if (tid < st) { sA[tid] += sA[tid + st]; sB[tid] += sB[tid + st]; }
    __syncthreads();
  }
  if (tid == 0) {
    float n = s_n;
    float denom = fmaxf(n - 1.0f, 1.0f);
    float stdv = sqrtf(sA[0] / denom) + sqrtf(sB[0] / denom);
    int lsum = 0;
    for (int q = 0; q < B; ++q) lsum += lens[q];
    float nb = (float)max(lsum, 1);
    float okf = ((p < lens[b]) && (n > 1.0f)) ? 1.0f : 0.0f;
    outp[idx] = okf * stdv / nb;
  }
}

__global__ __launch_bounds__(256) void cfl_final_kernel(const float* __restrict__ parts,
                                                        int n, float* __restrict__ out) {
  __shared__ float s_red[256];
  const int tid = threadIdx.x;
  float acc = 0.0f;
  for (int i = tid; i < n; i += 256) acc += parts[i];
  s_red[tid] = acc;
  __syncthreads();
  if (tid < 32) {
    float tot = wmma_sum256(s_red, tid);
    if (tid == 0) out[0] = tot;
  }
}

extern "C" void kernel_launch(void* const* d_in, const int* in_sizes, int n_in,
                              void* d_out, int out_size, void* d_ws, size_t ws_size,
                              hipStream_t stream) {
  (void)n_in; (void)out_size; (void)ws_size;
  const float* hm0 = (const float*)d_in[0];
  const float* hm1 = (const float*)d_in[1];
  const float* off_f = (const float*)d_in[2];
  const float* off_b = (const float*)d_in[3];
  const int* pts0 = (const int*)d_in[4];
  const int* pts1 = (const int*)d_in[5];
  const int* len0 = (const int*)d_in[6];
  const int* len1 = (const int*)d_in[7];

  const int B = in_sizes[0] / (kH * kW);
  const int P = in_sizes[4] / (B * 2);

  float* ws = (float*)d_ws;
  unsigned* norm_bits = (unsigned*)ws;
  float* main_parts = ws + 16;
  const int n_main = kBx * kBy * B;
  float* smooth_parts = main_parts + n_main;
  const int n_smooth = 2 * B * P;

  cfl_init_ws<<<1, 1, 0, stream>>>(ws);

  dim3 blk(kTile, kTile);
  dim3 grd(kBx, kBy, B);
  cfl_norm_kernel<<<grd, blk, 0, stream>>>(hm0, norm_bits);
  cfl_main_kernel<<<grd, blk, 0, stream>>>(hm0, hm1, off_f, off_b, norm_bits, main_parts);
  cfl_smooth_kernel<<<n_smooth, 128, 0, stream>>>(off_f, off_b, pts0, pts1, len0, len1,
                                                  B, P, smooth_parts);
  cfl_final_kernel<<<1, 256, 0, stream>>>(main_parts, n_main + n_smooth, (float*)d_out);
}